// FrequencyBands_76201309766078
// MI455X (gfx1250) — compile-verified
//
#include <hip/hip_runtime.h>
#include <math.h>

#define NUM_EXPERTS 16
#define FREQ_BINS   2097152   // 2^21
#define THREADS     256       // 8 wave32s per block
#define ELEMS_PER_THREAD 4    // one float4 (global_store_b128) per thread

// ---------------------------------------------------------------------------
// Kernel 1: tiny setup. One thread computes sigmoid(15 params), sorts the 17
// bounds (0 and 1 endpoints are already extremal since sigmoid in (0,1)),
// truncates to int32 bin indices, and emits int2{start,end} per expert.
// Cost: nanoseconds; irrelevant next to the 128 MiB store phase.
// ---------------------------------------------------------------------------
__global__ void FrequencyBands_compute_bounds(const float* __restrict__ bound_params,
                                              int2* __restrict__ se) {
    if (threadIdx.x != 0 || blockIdx.x != 0) return;

    float v[NUM_EXPERTS + 1];
    v[0]           = 0.0f;
    v[NUM_EXPERTS] = 1.0f;

    // Numerically stable sigmoid, matching jax.nn.sigmoid.
    #pragma unroll
    for (int i = 0; i < NUM_EXPERTS - 1; ++i) {
        float x = bound_params[i];
        float s;
        if (x >= 0.0f) {
            s = 1.0f / (1.0f + expf(-x));
        } else {
            float ex = expf(x);
            s = ex / (1.0f + ex);
        }
        v[i + 1] = s;
    }

    // Insertion sort of the 15 interior values (endpoints already sorted).
    for (int i = 2; i <= NUM_EXPERTS - 1; ++i) {
        float key = v[i];
        int j = i - 1;
        while (j >= 1 && v[j] > key) {
            v[j + 1] = v[j];
            --j;
        }
        v[j + 1] = key;
    }

    // indices = trunc(sorted * (FREQ_BINS-1)); starts = idx[0:16],
    // ends[e] = idx[e+1] for e<15, ends[15] = FREQ_BINS.
    int idx[NUM_EXPERTS + 1];
    #pragma unroll
    for (int i = 0; i <= NUM_EXPERTS; ++i) {
        idx[i] = (int)(v[i] * (float)(FREQ_BINS - 1));
    }
    #pragma unroll
    for (int e = 0; e < NUM_EXPERTS; ++e) {
        int2 b;
        b.x = idx[e];
        b.y = (e == NUM_EXPERTS - 1) ? FREQ_BINS : idx[e + 1];
        se[e] = b;
    }
}

// ---------------------------------------------------------------------------
// Kernel 2: the 128 MiB streaming write. blockIdx.y = expert, so the
// {start,end} fetch is block-uniform. Each lane produces 4 consecutive mask
// values with the branch-free unsigned range check and issues a single
// 16-byte store -> each wave32 emits one global_store_b128 covering 512
// contiguous bytes of the row. Store-bandwidth bound by construction; default
// RT temporal hint keeps the 128 MiB output resident in the 192 MB L2.
// ---------------------------------------------------------------------------
__global__ void __launch_bounds__(THREADS)
FrequencyBands_write_masks(const int2* __restrict__ se,
                           float* __restrict__ out) {
    const int expert = blockIdx.y;
    const int2 b = se[expert];                    // uniform per block
    const unsigned s   = (unsigned)b.x;
    const unsigned len = (unsigned)(b.y - b.x);   // end >= start (sorted)

    const unsigned t = blockIdx.x * THREADS + threadIdx.x; // float4 index in row
    const unsigned p = t * ELEMS_PER_THREAD;               // element position

    float4 val;
    val.x = ((p + 0u) - s) < len ? 1.0f : 0.0f;
    val.y = ((p + 1u) - s) < len ? 1.0f : 0.0f;
    val.z = ((p + 2u) - s) < len ? 1.0f : 0.0f;
    val.w = ((p + 3u) - s) < len ? 1.0f : 0.0f;

    float4* row = (float4*)(out + (size_t)expert * FREQ_BINS);
    row[t] = val;                                  // global_store_b128
}

extern "C" void kernel_launch(void* const* d_in, const int* in_sizes, int n_in,
                              void* d_out, int out_size, void* d_ws, size_t ws_size,
                              hipStream_t stream) {
    (void)in_sizes; (void)n_in; (void)out_size; (void)ws_size;

    const float* bound_params = (const float*)d_in[0]; // 15 floats
    float*       out          = (float*)d_out;         // 16 * 2097152 floats
    int2*        se           = (int2*)d_ws;           // 16 * {start,end}

    FrequencyBands_compute_bounds<<<1, 32, 0, stream>>>(bound_params, se);

    dim3 grid(FREQ_BINS / (ELEMS_PER_THREAD * THREADS), NUM_EXPERTS); // (2048, 16)
    FrequencyBands_write_masks<<<grid, THREADS, 0, stream>>>(se, out);
}